// EnergyConditionedEquivariantAtomAttention_34797825032839
// MI455X (gfx1250) — compile-verified
//
#include <hip/hip_runtime.h>
#include <math.h>

// ---------------- problem constants (from reference) ----------------
#define BATCH   16
#define NA      96
#define F_TOT   (BATCH*NA)      // 1536
#define NEDGE   256
#define BE_TOT  (BATCH*NEDGE)   // 4096
#define ATOMD   128
#define EDIMD   16
#define HIDD    128
#define LATD    128
#define RBFN    16
#define ZEMBD   32
#define E_ATT   1024
#define NODE_DIM 160            // 64 + 3*32
#define OUT_DIMV 80             // 32 + 3*16
#define WNUM     4608
#define QIN_DIM  144            // ATOM + EDIM  (already %4 == 0)
#define KIN_DIM  177            // ATOM + ZEMB + 1 + RBF
#define KIN_PAD  180            // padded to %4
#define WIN_DIM  49             // ZEMB + 1 + RBF
#define WIN_PAD  52             // padded to %4
#define CUTR     5.0f

typedef float v2f __attribute__((ext_vector_type(2)));
typedef float v8f __attribute__((ext_vector_type(8)));

__device__ __forceinline__ float siluf(float x) { return x / (1.0f + __expf(-x)); }

// =====================================================================
// Generic batched fp32 WMMA GEMM:  C[z] = act(scale * A[z]*B[z] + bias)
//   A: M x K row-major (lda), B: K x N row-major (ldb) or N x K if transB,
//   C: M x N row-major (ldc).  Each wave -> 16 x 64 strip (NT=4 tiles,
//   A-fragment reused across the 4 B-tiles). Block = 256 threads = 8 waves
//   stacked over 128 rows; all 8 waves share the same B fragments (WGP$ hits).
// REQUIREMENTS (guaranteed by caller):  M % 128 == 0,  K % 4 == 0,
//   lda/ldb even.  N may be ragged: B column accesses are address-clamped
//   and zero-selected (no EXEC divergence around the WMMA), stores skipped.
// Inner loop is phase-split (load all fragments, then 4 WMMAs) so the
// 5 loads per k-step are in flight together instead of load->wait->wmma.
// =====================================================================
#define GEMM_NT 4
__global__ void wmma_gemm_f32(const float* __restrict__ A,
                              const float* __restrict__ B,
                              const float* __restrict__ bias,
                              float* __restrict__ C,
                              int M, int K, int N,
                              int lda, int ldb, int ldc,
                              long sA, long sB, long sC,
                              float scale, int act, int transB)
{
    const int wave = threadIdx.x >> 5;
    const int lane = threadIdx.x & 31;
    const int half = lane >> 4;
    const int l16  = lane & 15;

    const int m0 = (blockIdx.y * 8 + wave) * 16;
    const int n0 = blockIdx.x * (16 * GEMM_NT);
    const long z = blockIdx.z;
    B += z * sB;
    C += z * sC;

    // A fragment pointer: this lane reads K pairs (half*2, half*2+1), step 4
    const float* Ap = A + z * sA + (long)(m0 + l16) * lda + half * 2;

    v8f acc[GEMM_NT];
#pragma unroll
    for (int t = 0; t < GEMM_NT; ++t)
#pragma unroll
        for (int i = 0; i < 8; ++i) acc[t][i] = 0.0f;

    int  col[GEMM_NT];
    bool cok[GEMM_NT];
#pragma unroll
    for (int t = 0; t < GEMM_NT; ++t) {
        col[t] = n0 + t * 16 + l16;
        cok[t] = col[t] < N;
    }

    if (!transB) {
        const float* bp[GEMM_NT];
#pragma unroll
        for (int t = 0; t < GEMM_NT; ++t)
            bp[t] = B + (cok[t] ? col[t] : 0) + (long)(half * 2) * ldb;
        const long step = 4L * (long)ldb;
#pragma unroll 4
        for (int k0 = 0; k0 < K; k0 += 4) {
            // ---- load phase: A frag + 4 B frags in flight together ----
            const v2f a = *reinterpret_cast<const v2f*>(Ap);
            Ap += 4;
            float bx[GEMM_NT], by[GEMM_NT];
#pragma unroll
            for (int t = 0; t < GEMM_NT; ++t) {
                bx[t] = bp[t][0];
                by[t] = bp[t][ldb];
                bp[t] += step;
            }
            // ---- compute phase: 4 WMMAs, no loads in between ----
#pragma unroll
            for (int t = 0; t < GEMM_NT; ++t) {
                v2f b;
                b.x = cok[t] ? bx[t] : 0.0f;
                b.y = cok[t] ? by[t] : 0.0f;
                acc[t] = __builtin_amdgcn_wmma_f32_16x16x4_f32(
                             false, a, false, b, (short)0, acc[t], false, false);
            }
        }
    } else {
        const float* bp[GEMM_NT];
#pragma unroll
        for (int t = 0; t < GEMM_NT; ++t)
            bp[t] = B + (long)(cok[t] ? col[t] : 0) * ldb + half * 2;
#pragma unroll 4
        for (int k0 = 0; k0 < K; k0 += 4) {
            // ---- load phase ----
            const v2f a = *reinterpret_cast<const v2f*>(Ap);
            Ap += 4;
            v2f br[GEMM_NT];
#pragma unroll
            for (int t = 0; t < GEMM_NT; ++t) {
                br[t] = *reinterpret_cast<const v2f*>(bp[t]);
                bp[t] += 4;
            }
            // ---- compute phase ----
#pragma unroll
            for (int t = 0; t < GEMM_NT; ++t) {
                v2f b;
                b.x = cok[t] ? br[t].x : 0.0f;
                b.y = cok[t] ? br[t].y : 0.0f;
                acc[t] = __builtin_amdgcn_wmma_f32_16x16x4_f32(
                             false, a, false, b, (short)0, acc[t], false, false);
            }
        }
    }

#pragma unroll
    for (int t = 0; t < GEMM_NT; ++t) {
        if (!cok[t]) continue;
        const float bv = bias ? bias[col[t]] : 0.0f;
#pragma unroll
        for (int i = 0; i < 8; ++i) {
            const int row = m0 + i + half * 8;      // M % 16 == 0 guaranteed
            float v = acc[t][i] * scale + bv;
            if (act) v = siluf(v);
            C[(long)row * ldc + col[t]] = v;
        }
    }
}

// =====================================================================
// Elementwise / gather kernels
// =====================================================================
__global__ void zero_kernel(float* p, int n)
{
    int i = blockIdx.x * blockDim.x + threadIdx.x;
    if (i < n) p[i] = 0.0f;
}

// Pad-copy a K x N weight matrix to Kp x N (zero rows K..Kp-1)
__global__ void padk_kernel(const float* __restrict__ src, float* __restrict__ dst,
                            int K, int Kp, int N)
{
    int idx = blockIdx.x * blockDim.x + threadIdx.x;
    if (idx >= Kp * N) return;
    int kp = idx / N, n = idx % N;
    dst[idx] = (kp < K) ? src[(long)kp * N + n] : 0.0f;
}

__global__ void scatter_kernel(const int* __restrict__ att_dst,
                               const float* __restrict__ att_dist,
                               const float* __restrict__ att_vec,
                               float* __restrict__ d_flat,
                               float* __restrict__ v_flat,
                               float* __restrict__ am)
{
    int j = blockIdx.x * blockDim.x + threadIdx.x;
    if (j >= E_ATT) return;
    int f = att_dst[j];
    d_flat[f] = att_dist[j];
    v_flat[f * 3 + 0] = att_vec[j * 3 + 0];
    v_flat[f * 3 + 1] = att_vec[j * 3 + 1];
    v_flat[f * 3 + 2] = att_vec[j * 3 + 2];
    am[f] = 1.0f;
}

// Per-atom feature prep: rbf, z-embedding gather, one-hot absorber,
// cutoff envelope, normalized direction (y1), and the concatenated
// (K-padded) inputs w_in (52) and atom_static (180).
__global__ void prep_kernel(const float* __restrict__ h,
                            const float* __restrict__ z_emb,
                            const int* __restrict__ zz,
                            const unsigned char* __restrict__ maskb,
                            const int* __restrict__ absorber,
                            const float* __restrict__ d_flat,
                            const float* __restrict__ v_flat,
                            const float* __restrict__ am,
                            float* __restrict__ attm,
                            float* __restrict__ envm,
                            float* __restrict__ y1,
                            float* __restrict__ w_in,
                            float* __restrict__ atom_static)
{
    __shared__ float szr[ZEMBD];
    __shared__ float srbf[RBFN];
    __shared__ float sd, sisabs;

    const int f = blockIdx.x;
    const int b = f / NA, n = f % NA;
    const int t = threadIdx.x;

    if (t == 0) {
        float d = d_flat[f];
        sd = d;
        float a  = am[f];
        float env = (d < CUTR) ? 0.5f * (__cosf(3.14159265358979f * d / CUTR) + 1.0f) : 0.0f;
        envm[f] = a * env;
        attm[f] = (a > 0.5f && maskb[f]) ? 1.0f : 0.0f;
        sisabs  = (n == absorber[b]) ? 1.0f : 0.0f;
        float vx = v_flat[f*3], vy = v_flat[f*3+1], vz = v_flat[f*3+2];
        float nrm = fmaxf(sqrtf(vx*vx + vy*vy + vz*vz), 1e-8f);
        const float s3 = 1.7320508075688772f;
        y1[f*3+0] = s3 * vx / nrm;
        y1[f*3+1] = s3 * vy / nrm;
        y1[f*3+2] = s3 * vz / nrm;
    }
    __syncthreads();
    if (t < RBFN) {
        const float delta = CUTR / (RBFN - 1);
        float mu = delta * (float)t;
        float dd = sd - mu;
        srbf[t] = __expf(-dd * dd / (2.0f * delta * delta));
    }
    if (t < ZEMBD) szr[t] = z_emb[(long)zz[f] * ZEMBD + t];
    __syncthreads();

    if (t < WIN_PAD) {
        float v = 0.0f;
        if (t < ZEMBD)         v = szr[t];
        else if (t == ZEMBD)   v = sisabs;
        else if (t < WIN_DIM)  v = srbf[t - ZEMBD - 1];
        w_in[(long)f * WIN_PAD + t] = v;
    }
    for (int c = t; c < KIN_PAD; c += blockDim.x) {
        float v = 0.0f;
        if (c < ATOMD)              v = h[(long)f * ATOMD + c];
        else if (c < ATOMD+ZEMBD)   v = szr[c - ATOMD];
        else if (c == ATOMD+ZEMBD)  v = sisabs;
        else if (c < KIN_DIM)       v = srbf[c - ATOMD - ZEMBD - 1];
        atom_static[(long)f * KIN_PAD + c] = v;
    }
}

// q_in[b,e,:] = concat(h[b, absorber[b], :], e_feat[e, :])
__global__ void qin_kernel(const float* __restrict__ h,
                           const float* __restrict__ e_feat,
                           const int* __restrict__ absorber,
                           float* __restrict__ q_in)
{
    int idx = blockIdx.x * blockDim.x + threadIdx.x;
    if (idx >= BE_TOT * QIN_DIM) return;
    int col = idx % QIN_DIM;
    int be  = idx / QIN_DIM;
    int b = be / NEDGE, e = be % NEDGE;
    float v;
    if (col < ATOMD) v = h[((long)(b * NA + absorber[b])) * ATOMD + col];
    else             v = e_feat[(long)e * EDIMD + (col - ATOMD)];
    q_in[idx] = v;
}

// Per-atom equivariant tensor contraction using the per-atom weights tpw.
__global__ void contract_kernel(const float* __restrict__ hf,
                                const float* __restrict__ tpw,
                                const float* __restrict__ y1,
                                const float* __restrict__ envm,
                                float* __restrict__ v_irrep)
{
    __shared__ float sh[NODE_DIM];   // x0 = sh[0..63], x1 = sh[64..159]
    __shared__ float sy[3];
    __shared__ float st1[32];
    __shared__ float ss01[16];

    const int f = blockIdx.x;
    const int t = threadIdx.x;
    const float* tp = tpw + (long)f * WNUM;

    for (int i = t; i < NODE_DIM; i += blockDim.x) sh[i] = hf[(long)f * NODE_DIM + i];
    if (t < 3) sy[t] = y1[f * 3 + t];
    __syncthreads();

    if (t < 32)
        st1[t] = sh[64 + t*3] * sy[0] + sh[64 + t*3 + 1] * sy[1] + sh[64 + t*3 + 2] * sy[2];
    if (t < 16) {
        float s = 0.0f;
        for (int i = 0; i < 64; ++i) s += sh[i] * tp[2048 + i*16 + t];   // w01
        ss01[t] = s;
    }
    __syncthreads();

    if (t < OUT_DIMV) {
        const float em    = envm[f];
        const float alpha = 0.10206207261596575f;  // 1/sqrt(96)
        const float cc    = 0.5773502691896258f;   // 1/sqrt(3)
        float out;
        if (t < 32) {
            float a0 = 0.0f, a1 = 0.0f;
            for (int i = 0; i < 64; ++i) a0 += sh[i]  * tp[i*32 + t];          // w00
            for (int i = 0; i < 32; ++i) a1 += st1[i] * tp[3584 + i*32 + t];   // w11
            out = alpha * (a0 + cc * a1);
        } else {
            int u = t - 32, o1 = u / 3, m = u % 3;
            float aw = 0.0f;
            for (int i = 0; i < 32; ++i) aw += sh[64 + i*3 + m] * tp[3072 + i*16 + o1]; // w10
            out = alpha * cc * (ss01[o1] * sy[m] + aw);
        }
        v_irrep[(long)f * OUT_DIMV + t] = out * em;
    }
}

// Masked softmax over n (96) with renormalization, in place on scores.
__global__ void softmax_kernel(float* __restrict__ scores,
                               const float* __restrict__ attm)
{
    __shared__ float red[128];
    const int be = blockIdx.x;          // b*NE + e
    const int b  = be / NEDGE;
    const int t  = threadIdx.x;
    float* row = scores + (long)be * NA;
    const float* mrow = attm + (long)b * NA;

    float m = (t < NA) ? mrow[t] : 0.0f;
    float s = (t < NA && m > 0.5f) ? row[t] : -1e30f;
    red[t] = s; __syncthreads();
    for (int off = 64; off >= 1; off >>= 1) {
        if (t < off) red[t] = fmaxf(red[t], red[t + off]);
        __syncthreads();
    }
    float mx = red[0]; __syncthreads();
    float p = (t < NA && m > 0.5f) ? __expf(s - mx) : 0.0f;
    red[t] = p; __syncthreads();
    for (int off = 64; off >= 1; off >>= 1) {
        if (t < off) red[t] += red[t + off];
        __syncthreads();
    }
    float sum = fmaxf(red[0], 1e-8f);
    if (t < NA) row[t] = p / sum;
}

// Expand gates (256x48) -> g_exp (256x80)
__global__ void gexp_kernel(const float* __restrict__ gates, float* __restrict__ gexp)
{
    int idx = blockIdx.x * blockDim.x + threadIdx.x;
    if (idx >= NEDGE * OUT_DIMV) return;
    int e = idx / OUT_DIMV, c = idx % OUT_DIMV;
    float v = (c < 32) ? gates[e*48 + c] : gates[e*48 + 32 + (c - 32)/3];
    gexp[idx] = v;
}

// Apply gating and build invariants: scal(32) + vector norms(16)
__global__ void inv_kernel(const float* __restrict__ oirr,
                           const float* __restrict__ gexp,
                           float* __restrict__ invb)
{
    int idx = blockIdx.x * blockDim.x + threadIdx.x;
    if (idx >= BE_TOT * 48) return;
    int be = idx / 48, c = idx % 48;
    int e = be % NEDGE;
    float r;
    if (c < 32) {
        r = oirr[(long)be*OUT_DIMV + c] * gexp[e*OUT_DIMV + c];
    } else {
        int j = c - 32;
        float g = gexp[e*OUT_DIMV + 32 + j*3];
        float s = 0.0f;
#pragma unroll
        for (int m = 0; m < 3; ++m) {
            float v = oirr[(long)be*OUT_DIMV + 32 + j*3 + m] * g;
            s += v * v;
        }
        r = sqrtf(s + 1e-12f);
    }
    invb[idx] = r;
}

// =====================================================================
// Host-side launch
// =====================================================================
static inline void launch_gemm(hipStream_t st,
                               const float* A, const float* B, const float* bias, float* C,
                               int M, int K, int N, int lda, int ldb, int ldc,
                               long sA, long sB, long sC,
                               float scale, int act, int transB, int Z)
{
    dim3 grid((N + 63) / 64, M / 128, Z);   // M is always a multiple of 128 here
    wmma_gemm_f32<<<grid, dim3(256), 0, st>>>(A, B, bias, C, M, K, N,
                                              lda, ldb, ldc, sA, sB, sC,
                                              scale, act, transB);
}

extern "C" void kernel_launch(void* const* d_in, const int* in_sizes, int n_in,
                              void* d_out, int out_size, void* d_ws, size_t ws_size,
                              hipStream_t stream)
{
    // ---- inputs (setup_inputs order) ----
    const float* h        = (const float*)d_in[0];
    const float* h_full   = (const float*)d_in[1];
    const float* e_feat   = (const float*)d_in[2];
    const float* att_dist = (const float*)d_in[3];
    const float* att_vec  = (const float*)d_in[4];
    const float* z_emb    = (const float*)d_in[5];
    const float* rw1 = (const float*)d_in[6];   const float* rb1 = (const float*)d_in[7];
    const float* rw2 = (const float*)d_in[8];   const float* rb2 = (const float*)d_in[9];
    const float* ew1 = (const float*)d_in[10];  const float* eb1 = (const float*)d_in[11];
    const float* ew2 = (const float*)d_in[12];  const float* eb2 = (const float*)d_in[13];
    const float* qw1 = (const float*)d_in[14];  const float* qb1 = (const float*)d_in[15];
    const float* qw2 = (const float*)d_in[16];  const float* qb2 = (const float*)d_in[17];
    const float* qw3 = (const float*)d_in[18];  const float* qb3 = (const float*)d_in[19];
    const float* kw1 = (const float*)d_in[20];  const float* kb1 = (const float*)d_in[21];
    const float* kw2 = (const float*)d_in[22];  const float* kb2 = (const float*)d_in[23];
    const float* kw3 = (const float*)d_in[24];  const float* kb3 = (const float*)d_in[25];
    const float* ow1 = (const float*)d_in[26];  const float* ob1 = (const float*)d_in[27];
    const float* ow2 = (const float*)d_in[28];  const float* ob2 = (const float*)d_in[29];
    const float* ow3 = (const float*)d_in[30];  const float* ob3 = (const float*)d_in[31];
    const int*   zz       = (const int*)d_in[32];
    const unsigned char* maskb = (const unsigned char*)d_in[33];   // jnp bool = 1 byte
    const int*   absorber = (const int*)d_in[34];
    const int*   att_dst  = (const int*)d_in[35];

    // ---- workspace carve (floats) ----
    float* W = (float*)d_ws;
    size_t off = 0;
    auto alloc = [&](size_t n) { float* p = W + off; off += n; return p; };
    float* d_flat      = alloc(F_TOT);
    float* v_flat      = alloc(3 * F_TOT);
    float* am          = alloc(F_TOT);
    float* attm        = alloc(F_TOT);
    float* envm        = alloc(F_TOT);
    float* y1b         = alloc(3 * F_TOT);
    float* w_in        = alloc((size_t)F_TOT * WIN_PAD);
    float* atom_static = alloc((size_t)F_TOT * KIN_PAD);
    float* q_in        = alloc((size_t)BE_TOT * QIN_DIM);
    float* bufB        = alloc((size_t)BE_TOT * HIDD);
    float* bufC        = alloc((size_t)BE_TOT * HIDD);
    float* tpw         = alloc((size_t)F_TOT * WNUM);
    float* v_irrep     = alloc((size_t)F_TOT * OUT_DIMV);
    float* qbuf        = alloc((size_t)BE_TOT * LATD);
    float* kbuf        = alloc((size_t)F_TOT * LATD);
    float* scores      = alloc((size_t)BATCH * NEDGE * NA);
    float* oirr        = alloc((size_t)BE_TOT * OUT_DIMV);
    float* gates       = alloc((size_t)NEDGE * 48);
    float* gexp        = alloc((size_t)NEDGE * OUT_DIMV);
    float* invb        = alloc((size_t)BE_TOT * 48);
    float* rw1p        = alloc((size_t)WIN_PAD * HIDD);   // K-padded weights
    float* kw1p        = alloc((size_t)KIN_PAD * HIDD);
    (void)ws_size; (void)n_in; (void)in_sizes; (void)out_size;

    float* outp = (float*)d_out;

    // 0) K-pad the two odd-K weight matrices
    padk_kernel<<<(WIN_PAD*HIDD + 255)/256, 256, 0, stream>>>(rw1, rw1p, WIN_DIM, WIN_PAD, HIDD);
    padk_kernel<<<(KIN_PAD*HIDD + 255)/256, 256, 0, stream>>>(kw1, kw1p, KIN_DIM, KIN_PAD, HIDD);

    // 1) zero edge scatter targets (d_flat, v_flat, am contiguous: 5F)
    zero_kernel<<<(5 * F_TOT + 255) / 256, 256, 0, stream>>>(d_flat, 5 * F_TOT);
    // 2) scatter attention edges
    scatter_kernel<<<(E_ATT + 255) / 256, 256, 0, stream>>>(att_dst, att_dist, att_vec,
                                                            d_flat, v_flat, am);
    // 3) per-atom features (writes K-padded rows)
    prep_kernel<<<F_TOT, 64, 0, stream>>>(h, z_emb, zz, maskb, absorber,
                                          d_flat, v_flat, am, attm, envm, y1b,
                                          w_in, atom_static);
    // 4) q_in concat
    qin_kernel<<<(BE_TOT * QIN_DIM + 255) / 256, 256, 0, stream>>>(h, e_feat, absorber, q_in);

    // 5) gate MLP: e_feat(256x16) -> silu(128) -> 48
    launch_gemm(stream, e_feat, ew1, eb1, bufB, NEDGE, EDIMD, HIDD, EDIMD, HIDD, HIDD,
                0, 0, 0, 1.0f, 1, 0, 1);
    launch_gemm(stream, bufB, ew2, eb2, gates, NEDGE, HIDD, 48, HIDD, 48, 48,
                0, 0, 0, 1.0f, 0, 0, 1);
    gexp_kernel<<<(NEDGE * OUT_DIMV + 255) / 256, 256, 0, stream>>>(gates, gexp);

    // 6) radial net: w_in(1536x52p) -> silu(128) -> tpw(4608)   [dominant GEMM]
    launch_gemm(stream, w_in, rw1p, rb1, bufB, F_TOT, WIN_PAD, HIDD, WIN_PAD, HIDD, HIDD,
                0, 0, 0, 1.0f, 1, 0, 1);
    launch_gemm(stream, bufB, rw2, rb2, tpw, F_TOT, HIDD, WNUM, HIDD, WNUM, WNUM,
                0, 0, 0, 1.0f, 0, 0, 1);

    // 7) equivariant contraction -> v_irrep (envelope applied)
    contract_kernel<<<F_TOT, 96, 0, stream>>>(h_full, tpw, y1b, envm, v_irrep);

    // 8) q MLP: q_in(4096x144) -> 128 -> 128 -> 128
    launch_gemm(stream, q_in, qw1, qb1, bufB, BE_TOT, QIN_DIM, HIDD, QIN_DIM, HIDD, HIDD,
                0, 0, 0, 1.0f, 1, 0, 1);
    launch_gemm(stream, bufB, qw2, qb2, bufC, BE_TOT, HIDD, HIDD, HIDD, HIDD, HIDD,
                0, 0, 0, 1.0f, 1, 0, 1);
    launch_gemm(stream, bufC, qw3, qb3, qbuf, BE_TOT, HIDD, LATD, HIDD, LATD, LATD,
                0, 0, 0, 1.0f, 0, 0, 1);

    // 9) k MLP: atom_static(1536x180p) -> 128 -> 128 -> 128
    launch_gemm(stream, atom_static, kw1p, kb1, bufB, F_TOT, KIN_PAD, HIDD, KIN_PAD, HIDD, HIDD,
                0, 0, 0, 1.0f, 1, 0, 1);
    launch_gemm(stream, bufB, kw2, kb2, bufC, F_TOT, HIDD, HIDD, HIDD, HIDD, HIDD,
                0, 0, 0, 1.0f, 1, 0, 1);
    launch_gemm(stream, bufC, kw3, kb3, kbuf, F_TOT, HIDD, LATD, HIDD, LATD, LATD,
                0, 0, 0, 1.0f, 0, 0, 1);

    // 10) scores[b] = scale * q[b](256x128) * k[b]^T(128x96)   (batched, transB)
    const float score_scale = 1.0f / (4.0f * sqrtf(32.0f));   // mean over heads * HD^-0.5
    launch_gemm(stream, qbuf, kbuf, nullptr, scores, NEDGE, LATD, NA, LATD, LATD, NA,
                (long)NEDGE * LATD, (long)NA * LATD, (long)NEDGE * NA,
                score_scale, 0, 1, BATCH);

    // 11) masked softmax + renorm (in place)
    softmax_kernel<<<BE_TOT, 128, 0, stream>>>(scores, attm);

    // 12) out_irrep[b] = attn[b](256x96) * v_irrep[b](96x80)   (batched)
    launch_gemm(stream, scores, v_irrep, nullptr, oirr, NEDGE, NA, OUT_DIMV, NA, OUT_DIMV, OUT_DIMV,
                (long)NEDGE * NA, (long)NA * OUT_DIMV, (long)NEDGE * OUT_DIMV,
                1.0f, 0, 0, BATCH);

    // 13) gating + invariants
    inv_kernel<<<(BE_TOT * 48 + 255) / 256, 256, 0, stream>>>(oirr, gexp, invb);

    // 14) output MLP: inv(4096x48) -> 128 -> 128 -> 128 (to d_out)
    launch_gemm(stream, invb, ow1, ob1, bufB, BE_TOT, 48, HIDD, 48, HIDD, HIDD,
                0, 0, 0, 1.0f, 1, 0, 1);
    launch_gemm(stream, bufB, ow2, ob2, bufC, BE_TOT, HIDD, HIDD, HIDD, HIDD, HIDD,
                0, 0, 0, 1.0f, 1, 0, 1);
    launch_gemm(stream, bufC, ow3, ob3, outp, BE_TOT, HIDD, LATD, HIDD, LATD, LATD,
                0, 0, 0, 1.0f, 0, 0, 1);
}